// LinearAttention_25898652795099
// MI455X (gfx1250) — compile-verified
//
#include <hip/hip_runtime.h>
#include <hip/hip_bf16.h>
#include <stdint.h>

// ---------------------------------------------------------------------------
// Linear attention (ELU feature map), MI455X / gfx1250, wave32.
// Bandwidth-bound (157 MB moved vs 1.3 GFLOP) -> stream-once design,
// matrix work done with full-precision V_WMMA_F32_16X16X4_F32.
// ---------------------------------------------------------------------------

typedef float v2f __attribute__((ext_vector_type(2)));
typedef float v8f __attribute__((ext_vector_type(8)));

#define NH      8
#define BNH     64          // B*nh
#define LQ      4800
#define SK      4800
#define DD      32
#define NCHUNK  (SK / 4)    // 1200 K-dim chunks of 4 for 16x16x4 WMMA
#define NTILES  (LQ / 16)   // 300 q-row tiles per head
#define TBLK    38          // ceil(300/8) tile-blocks per head
#define EPS_F   1e-6f

__device__ __forceinline__ float elu1(float x) {
    // elu(x)+1 = x+1 (x>0) else exp(x)
    return x > 0.0f ? x + 1.0f : __expf(x);
}

// ---------------------------------------------------------------------------
// Phase 1: per (head, split) accumulate partial kv[32x32] = phi(K)^T (V/S)
// and partial k_sum[32], streaming S. WMMA f32 16x16x4, K-dim = 4 s-rows.
// Fragment layouts per CDNA5 ISA 7.12.2 (32-bit A 16x4 / B 4x16 / C 16x16).
// ---------------------------------------------------------------------------
__global__ __launch_bounds__(256) void la_phase1(
    const float* __restrict__ k, const float* __restrict__ v,
    const uint8_t* __restrict__ kv_mask,
    float* __restrict__ ws_kv, float* __restrict__ ws_ks, int nsplit)
{
    __shared__ float red_kv[8][1024];   // per-wave partial kv, 32 KB
    __shared__ float red_ks[8][64];     // per-wave partial k_sum (2 halves x 32 lanes)

    const int bh    = blockIdx.x / nsplit;
    const int split = blockIdx.x % nsplit;
    const int b     = bh / NH;
    const int wave  = threadIdx.x >> 5;
    const int lane  = threadIdx.x & 31;
    const int m     = lane & 15;              // A-row (d) / B-col (e) index
    const int off   = (lane < 16) ? 0 : 2;    // K-slot base per ISA layout

    const float invS = 1.0f / (float)SK;
    const float*   kp = k + (size_t)bh * SK * DD;
    const float*   vp = v + (size_t)bh * SK * DD;
    const uint8_t* km = kv_mask + (size_t)b * SK;

    v8f acc00 = {}, acc01 = {}, acc10 = {}, acc11 = {};
    float ks0 = 0.0f, ks1 = 0.0f;

    for (int c = split * 8 + wave; c < NCHUNK; c += nsplit * 8) {
        const int sA = c * 4 + off;
        const int sB = sA + 1;
        const float mA = km[sA] ? 1.0f : 0.0f;
        const float mB = km[sB] ? 1.0f : 0.0f;

        // A fragments: phi(K)^T tiles, d in [0,16) and [16,32)
        const float ka0 = elu1(kp[sA * DD + m])      * mA;
        const float ka1 = elu1(kp[sB * DD + m])      * mB;
        const float kb0 = elu1(kp[sA * DD + 16 + m]) * mA;
        const float kb1 = elu1(kp[sB * DD + 16 + m]) * mB;
        ks0 += ka0 + ka1;
        ks1 += kb0 + kb1;

        // B fragments: (V/S) tiles, e in [0,16) and [16,32)
        const float va0 = vp[sA * DD + m]      * (mA * invS);
        const float va1 = vp[sB * DD + m]      * (mB * invS);
        const float vb0 = vp[sA * DD + 16 + m] * (mA * invS);
        const float vb1 = vp[sB * DD + 16 + m] * (mB * invS);

        v2f A0 = {ka0, ka1}, A1 = {kb0, kb1};
        v2f B0 = {va0, va1}, B1 = {vb0, vb1};
        acc00 = __builtin_amdgcn_wmma_f32_16x16x4_f32(false, A0, false, B0, (short)0, acc00, false, false);
        acc01 = __builtin_amdgcn_wmma_f32_16x16x4_f32(false, A0, false, B1, (short)0, acc01, false, false);
        acc10 = __builtin_amdgcn_wmma_f32_16x16x4_f32(false, A1, false, B0, (short)0, acc10, false, false);
        acc11 = __builtin_amdgcn_wmma_f32_16x16x4_f32(false, A1, false, B1, (short)0, acc11, false, false);
    }

    // Deterministic cross-wave reduction via LDS (no float atomics).
    const int rbase = (lane >= 16) ? 8 : 0;     // C/D layout: VGPR r -> row r or r+8
#pragma unroll
    for (int r = 0; r < 8; ++r) {
        const int row = r + rbase;
        red_kv[wave][row * 32 + m]             = acc00[r];
        red_kv[wave][row * 32 + 16 + m]        = acc01[r];
        red_kv[wave][(16 + row) * 32 + m]      = acc10[r];
        red_kv[wave][(16 + row) * 32 + 16 + m] = acc11[r];
    }
    red_ks[wave][lane]      = ks0;   // d = lane%16
    red_ks[wave][32 + lane] = ks1;   // d = 16 + lane%16
    __syncthreads();

    float* okv = ws_kv + (size_t)(bh * nsplit + split) * 1024;
#pragma unroll
    for (int i = threadIdx.x; i < 1024; i += 256) {
        float s = 0.0f;
#pragma unroll
        for (int w = 0; w < 8; ++w) s += red_kv[w][i];
        okv[i] = s;
    }
    if (threadIdx.x < 32) {
        const int d  = threadIdx.x;
        const int i0 = (d < 16) ? d        : (d + 16);
        const int i1 = (d < 16) ? (d + 16) : (d + 32);
        float s = 0.0f;
#pragma unroll
        for (int w = 0; w < 8; ++w) s += red_ks[w][i0] + red_ks[w][i1];
        ws_ks[(size_t)(bh * nsplit + split) * 32 + d] = s;
    }
}

// ---------------------------------------------------------------------------
// Phase 2: y = (phi(Q) @ kv) * S / (phi(Q)·k_sum + eps).
// One 16-row q tile per wave; kv/k_sum staged into LDS per block (split-sum
// happens here, served from the 192 MB L2).
// ---------------------------------------------------------------------------
__global__ __launch_bounds__(256) void la_phase2(
    const float* __restrict__ q, const uint8_t* __restrict__ q_mask,
    const float* __restrict__ ws_kv, const float* __restrict__ ws_ks,
    float* __restrict__ out, int nsplit)
{
    __shared__ float s_kv[1024];
    __shared__ float s_ks[32];
    __shared__ float s_q[8][16 * 32];   // per-wave staged phi(q) tile

    const int bh   = blockIdx.x / TBLK;
    const int tblk = blockIdx.x % TBLK;
    const int b    = bh / NH;
    const int wave = threadIdx.x >> 5;
    const int lane = threadIdx.x & 31;
    const int m    = lane & 15;
    const int off  = (lane < 16) ? 0 : 2;
    const int tile = tblk * 8 + wave;
    const int row0 = tile * 16;

    // Stage kv / k_sum (sum split partials).
    for (int i = threadIdx.x; i < 1024; i += 256) {
        float s = 0.0f;
        for (int p = 0; p < nsplit; ++p) s += ws_kv[(size_t)(bh * nsplit + p) * 1024 + i];
        s_kv[i] = s;
    }
    if (threadIdx.x < 32) {
        float s = 0.0f;
        for (int p = 0; p < nsplit; ++p) s += ws_ks[(size_t)(bh * nsplit + p) * 32 + threadIdx.x];
        s_ks[threadIdx.x] = s;
    }
    // Stage phi(q)*mask tile: 64B per lane, coalesced float4 loads.
    if (tile < NTILES) {
        const float* qp  = q + ((size_t)bh * LQ + row0) * DD;
        const int    row = lane >> 1;
        const float  msk = q_mask[(size_t)b * LQ + row0 + row] ? 1.0f : 0.0f;
        const float4* src = (const float4*)(qp + lane * 16);
        float*        dst = &s_q[wave][lane * 16];
#pragma unroll
        for (int i = 0; i < 4; ++i) {
            float4 f = src[i];
            dst[i * 4 + 0] = elu1(f.x) * msk;
            dst[i * 4 + 1] = elu1(f.y) * msk;
            dst[i * 4 + 2] = elu1(f.z) * msk;
            dst[i * 4 + 3] = elu1(f.w) * msk;
        }
    }
    __syncthreads();
    if (tile >= NTILES) return;   // wave-uniform: EXEC all-ones in live path

    v8f acc0 = {}, acc1 = {};
    float den = 0.0f;
#pragma unroll
    for (int c = 0; c < 8; ++c) {
        const int col = 4 * c + off;                       // even -> 8B aligned
        const float2 a  = *(const float2*)&s_q[wave][m * 32 + col];
        const float2 ks = *(const float2*)&s_ks[col];
        den += a.x * ks.x + a.y * ks.y;                    // partial q·k_sum
        v2f A  = {a.x, a.y};
        v2f B0 = {s_kv[col * 32 + m],      s_kv[(col + 1) * 32 + m]};
        v2f B1 = {s_kv[col * 32 + 16 + m], s_kv[(col + 1) * 32 + 16 + m]};
        acc0 = __builtin_amdgcn_wmma_f32_16x16x4_f32(false, A, false, B0, (short)0, acc0, false, false);
        acc1 = __builtin_amdgcn_wmma_f32_16x16x4_f32(false, A, false, B1, (short)0, acc1, false, false);
    }
    // lanes l and l^16 hold the two halves of row (l%16)'s dot product
    den += __shfl_xor(den, 16, 32);
    const float z = (float)SK / (den + EPS_F);   // S * 1/(den+eps); 1/S folded in kv

    float* op = out + ((size_t)bh * LQ + row0) * DD;
    const int rbase = (lane >= 16) ? 8 : 0;
#pragma unroll
    for (int r = 0; r < 8; ++r) {
        const float zr  = __shfl(z, (lane < 16) ? r : (r + 8), 32);
        const int   row = r + rbase;
        op[row * DD + m]      = acc0[r] * zr;
        op[row * DD + 16 + m] = acc1[r] * zr;
    }
}

// ---------------------------------------------------------------------------
extern "C" void kernel_launch(void* const* d_in, const int* in_sizes, int n_in,
                              void* d_out, int out_size, void* d_ws, size_t ws_size,
                              hipStream_t stream) {
    const float*   q  = (const float*)d_in[0];
    const float*   k  = (const float*)d_in[1];
    const float*   v  = (const float*)d_in[2];
    const uint8_t* qm = (const uint8_t*)d_in[3];   // bool mask [B, L]
    const uint8_t* km = (const uint8_t*)d_in[4];   // bool mask [B, S]
    float*         o  = (float*)d_out;

    int nsplit = 8;   // S-splits per head in phase 1 -> 512 blocks (fill WGPs)
    while (nsplit > 1 && (size_t)BNH * nsplit * (1024 + 32) * sizeof(float) > ws_size)
        nsplit >>= 1;

    float* ws_kv = (float*)d_ws;
    float* ws_ks = ws_kv + (size_t)BNH * nsplit * 1024;

    la_phase1<<<BNH * nsplit, 256, 0, stream>>>(k, v, km, ws_kv, ws_ks, nsplit);
    la_phase2<<<BNH * TBLK, 256, 0, stream>>>(q, qm, ws_kv, ws_ks, o, nsplit);
}